// Invert4_10_20684562497572
// MI455X (gfx1250) — compile-verified
//
#include <hip/hip_runtime.h>

#define TPB   256
#define PIPE  4
#define MAX_BLOCKS 8192

typedef __attribute__((ext_vector_type(4))) float floatx4;
typedef __attribute__((ext_vector_type(4))) int   intx4;

#if defined(__HIP_DEVICE_COMPILE__)
  #if __has_builtin(__builtin_amdgcn_global_load_async_to_lds_b128)
    #define USE_ASYNC 1
  #else
    #define USE_ASYNC 0
  #endif
  #if __has_builtin(__builtin_amdgcn_s_wait_asynccnt)
    #define WAIT_ASYNC(n) __builtin_amdgcn_s_wait_asynccnt(n)
  #else
    #define WAIT_ASYNC(n) asm volatile("s_wait_asynccnt " #n ::: "memory")
  #endif
  #define WAIT_DS0() asm volatile("s_wait_dscnt 0x0" ::: "memory")
#else
  #define USE_ASYNC 0
  #define WAIT_ASYNC(n) ((void)0)
  #define WAIT_DS0()    ((void)0)
#endif

#define AS1 __attribute__((address_space(1)))
#define AS3 __attribute__((address_space(3)))

// ---- elementwise 16-step spike scan; exact-match with reference:
// v -= z*h (z in {0,1} -> product exact, fma == mul+sub);
// spike((v-T)/(|v|+1)) > 0  <=>  v > T  (denominator strictly positive);
// out += z*d (exact).
__device__ __forceinline__ float snn_elem(float xx, const float* hr,
                                          const float* dr, const float* Tr) {
  float sgn = (xx > 0.0f) ? 1.0f : ((xx < 0.0f) ? -1.0f : 0.0f);
  float v = __builtin_fabsf(xx);
  float z = 0.0f, o = 0.0f;
#pragma unroll
  for (int t = 0; t < 16; ++t) {
    v = __builtin_fmaf(-hr[t], z, v);   // v -= z*h[t]
    z = (v > Tr[t]) ? 1.0f : 0.0f;      // spike sign test (division eliminated)
    o = __builtin_fmaf(z, dr[t], o);    // out += z*d[t]
  }
  return o * sgn;
}

__device__ __forceinline__ floatx4 snn_vec4(floatx4 xv, const float* hr,
                                            const float* dr, const float* Tr) {
  floatx4 o;
#pragma unroll
  for (int c = 0; c < 4; ++c) o[c] = snn_elem(xv[c], hr, dr, Tr);
  return o;
}

__device__ __forceinline__ void wait_async_upto(int a) {
  switch (a) {
    case 0:  WAIT_ASYNC(0); break;
    case 1:  WAIT_ASYNC(1); break;
    case 2:  WAIT_ASYNC(2); break;
    default: WAIT_ASYNC(3); break;
  }
}

__global__ void __launch_bounds__(TPB)
snn_stream_kernel(const float* __restrict__ x, const float* __restrict__ h,
                  const float* __restrict__ d, const float* __restrict__ T,
                  float* __restrict__ out, long long n4) {
  // Uniform param loads -> SGPRs (pointer uniform, indices compile-time).
  float hr[16], dr[16], Tr[16];
#pragma unroll
  for (int t = 0; t < 16; ++t) { hr[t] = h[t]; dr[t] = d[t]; Tr[t] = T[t]; }

  const int tid = threadIdx.x;
  const long long ntiles = (n4 + TPB - 1) / TPB;
  const long long stride = (long long)gridDim.x;
  const floatx4* __restrict__ x4 = (const floatx4*)x;
  floatx4* __restrict__ out4 = (floatx4*)out;

#if USE_ASYNC
  __shared__ floatx4 buf[PIPE][TPB];

  // Prime the pipeline: up to PIPE tiles in flight per wave (ASYNCcnt-tracked,
  // zero VGPR destinations held).
#pragma unroll
  for (int k = 0; k < PIPE; ++k) {
    long long tk = (long long)blockIdx.x + (long long)k * stride;
    if (tk < ntiles) {
      long long idx = tk * TPB + tid;
      if (idx < n4) {
        __builtin_amdgcn_global_load_async_to_lds_b128(
            (AS1 intx4*)(x4 + idx), (AS3 intx4*)&buf[k][tid], 0, 0);
      }
    }
  }

  int slot = 0;
  for (long long i = (long long)blockIdx.x; i < ntiles; i += stride) {
    // Async loads complete in order: wait until <= (#issued after this tile).
    long long after = (ntiles - 1 - i) / stride;        // future tiles for this block
    int a = (after >= (long long)(PIPE - 1)) ? (PIPE - 1) : (int)after;
    wait_async_upto(a);

    long long idx = i * TPB + tid;
    if (idx < n4) {
      floatx4 r = buf[slot][tid];                       // ds_load_b128 (own slot)
      floatx4 o = snn_vec4(r, hr, dr, Tr);
      __builtin_nontemporal_store(o, out4 + idx);       // NT: don't thrash L2
    }

    // WAR guard: ensure our ds_load drained before the async engine overwrites
    // this LDS slot with tile i + PIPE*stride.
    WAIT_DS0();
    long long nt = i + (long long)PIPE * stride;
    if (nt < ntiles) {
      long long nidx = nt * TPB + tid;
      if (nidx < n4) {
        __builtin_amdgcn_global_load_async_to_lds_b128(
            (AS1 intx4*)(x4 + nidx), (AS3 intx4*)&buf[slot][tid], 0, 0);
      }
    }
    slot = (slot + 1 == PIPE) ? 0 : slot + 1;
  }
#else
  // Fallback: direct nontemporal streaming + explicit prefetch
  // (emits global_prefetch_b8 on gfx1250).
  const long long tstride = stride * TPB;
  for (long long i = (long long)blockIdx.x * TPB + tid; i < n4; i += tstride) {
    if (i + tstride < n4) __builtin_prefetch(x4 + i + tstride, 0, 0);
    floatx4 r = __builtin_nontemporal_load(x4 + i);
    floatx4 o = snn_vec4(r, hr, dr, Tr);
    __builtin_nontemporal_store(o, out4 + i);
  }
#endif
}

// Scalar tail for n % 4 (not hit for this problem size, kept for correctness).
__global__ void snn_tail_kernel(const float* __restrict__ x, const float* __restrict__ h,
                                const float* __restrict__ d, const float* __restrict__ T,
                                float* __restrict__ out, long long start, long long n) {
  float hr[16], dr[16], Tr[16];
#pragma unroll
  for (int t = 0; t < 16; ++t) { hr[t] = h[t]; dr[t] = d[t]; Tr[t] = T[t]; }
  long long i = start + (long long)blockIdx.x * blockDim.x + threadIdx.x;
  if (i < n) out[i] = snn_elem(x[i], hr, dr, Tr);
}

extern "C" void kernel_launch(void* const* d_in, const int* in_sizes, int n_in,
                              void* d_out, int out_size, void* d_ws, size_t ws_size,
                              hipStream_t stream) {
  const float* x = (const float*)d_in[0];
  const float* h = (const float*)d_in[1];
  const float* d = (const float*)d_in[2];
  const float* T = (const float*)d_in[3];
  float* out = (float*)d_out;

  long long n  = (long long)out_size;   // == in_sizes[0] == 8*4096*4096
  long long n4 = n / 4;

  if (n4 > 0) {
    long long ntiles = (n4 + TPB - 1) / TPB;
    int grid = (int)((ntiles < (long long)MAX_BLOCKS) ? ntiles : (long long)MAX_BLOCKS);
    if (grid < 1) grid = 1;
    snn_stream_kernel<<<grid, TPB, 0, stream>>>(x, h, d, T, out, n4);
  }
  long long rem = n - n4 * 4;
  if (rem > 0) {
    snn_tail_kernel<<<1, 64, 0, stream>>>(x, h, d, T, out, n4 * 4, n);
  }
}